// HyperNet2_51831665328138
// MI455X (gfx1250) — compile-verified
//
#include <hip/hip_runtime.h>
#include <math.h>

#define B_N 100
#define EPS_BN 1e-5f

typedef float v2f __attribute__((ext_vector_type(2)));
typedef float v8f __attribute__((ext_vector_type(8)));

// ---------------- generic 3x3 SAME conv: in [B,CIN,S,S] -> out [B,COUT,S,S] ----------------
template<int CIN, int COUT, int S>
__global__ void convk(const float* __restrict__ in, const float* __restrict__ w,
                      const float* __restrict__ bias, float* __restrict__ out) {
    int t = blockIdx.x * blockDim.x + threadIdx.x;
    if (t >= B_N * COUT * S * S) return;
    int p = t % (S * S);
    int rest = t / (S * S);
    int co = rest % COUT;
    int b  = rest / COUT;
    int i = p / S, j = p % S;
    float s = bias[co];
    for (int ci = 0; ci < CIN; ++ci) {
        const float* inp = in + (size_t)(b * CIN + ci) * S * S;
        const float* wp  = w + (size_t)(co * CIN + ci) * 9;
        #pragma unroll
        for (int dy = -1; dy <= 1; ++dy) {
            int ii = i + dy;
            if (ii < 0 || ii >= S) continue;
            #pragma unroll
            for (int dx = -1; dx <= 1; ++dx) {
                int jj = j + dx;
                if (jj < 0 || jj >= S) continue;
                s += wp[(dy + 1) * 3 + (dx + 1)] * inp[ii * S + jj];
            }
        }
    }
    out[t] = s;
}

// ---------------- per-channel batch stats (mean, invstd) ; deterministic tree reduce -------
template<int C, int S>
__global__ void stats_kernel(const float* __restrict__ in, float* __restrict__ stats) {
    const int c = blockIdx.x;
    const int tid = threadIdx.x;
    const int N = B_N * S * S;
    float s = 0.f, s2 = 0.f;
    for (int idx = tid; idx < N; idx += 256) {
        int b = idx / (S * S);
        int p = idx % (S * S);
        float v = in[(size_t)(b * C + c) * S * S + p];
        s += v; s2 += v * v;
    }
    __shared__ float sh[256];
    __shared__ float sh2[256];
    sh[tid] = s; sh2[tid] = s2;
    __syncthreads();
    for (int off = 128; off > 0; off >>= 1) {
        if (tid < off) { sh[tid] += sh[tid + off]; sh2[tid] += sh2[tid + off]; }
        __syncthreads();
    }
    if (tid == 0) {
        float mean = sh[0] / (float)N;
        float var  = sh2[0] / (float)N - mean * mean;
        stats[c * 2 + 0] = mean;
        stats[c * 2 + 1] = rsqrtf(var + EPS_BN);
    }
}

// ---------------- fused BN(batch stats)+ReLU+maxpool2: [B,C,S,S] -> [B,C,S/2,S/2] ----------
template<int C, int S>
__global__ void bnpool_kernel(const float* __restrict__ in, const float* __restrict__ stats,
                              const float* __restrict__ g, const float* __restrict__ beta,
                              float* __restrict__ out) {
    const int SO = S / 2;
    int t = blockIdx.x * blockDim.x + threadIdx.x;
    if (t >= B_N * C * SO * SO) return;
    int p = t % (SO * SO);
    int rest = t / (SO * SO);
    int c = rest % C;
    int b = rest / C;
    int i = p / SO, j = p % SO;
    float mean = stats[c * 2 + 0], inv = stats[c * 2 + 1];
    float gg = g[c], bb = beta[c];
    const float* inp = in + (size_t)(b * C + c) * S * S;
    float m = -1e30f;
    #pragma unroll
    for (int dy = 0; dy < 2; ++dy)
        #pragma unroll
        for (int dx = 0; dx < 2; ++dx) {
            float v = inp[(2 * i + dy) * S + (2 * j + dx)];
            v = (v - mean) * inv * gg + bb;
            m = fmaxf(m, v);
        }
    out[t] = fmaxf(m, 0.f);   // relu(max) == max(relu) (monotone)
}

// ---------------- conv4 (4->2, 4x4) + relu + flatten to feat [B,32] ------------------------
__global__ void conv4_kernel(const float* __restrict__ in, const float* __restrict__ w,
                             const float* __restrict__ bias, float* __restrict__ feat) {
    int t = blockIdx.x * blockDim.x + threadIdx.x;
    if (t >= B_N * 2 * 16) return;
    int p = t & 15;
    int rest = t >> 4;
    int co = rest & 1;
    int b  = rest >> 1;
    int i = p >> 2, j = p & 3;
    float s = bias[co];
    for (int ci = 0; ci < 4; ++ci) {
        const float* inp = in + (size_t)(b * 4 + ci) * 16;
        const float* wp  = w + (size_t)(co * 4 + ci) * 9;
        #pragma unroll
        for (int dy = -1; dy <= 1; ++dy) {
            int ii = i + dy;
            if (ii < 0 || ii >= 4) continue;
            #pragma unroll
            for (int dx = -1; dx <= 1; ++dx) {
                int jj = j + dx;
                if (jj < 0 || jj >= 4) continue;
                s += wp[(dy + 1) * 3 + (dx + 1)] * inp[ii * 4 + jj];
            }
        }
    }
    feat[b * 32 + co * 16 + p] = fmaxf(s, 0.f);
}

// ---------------- hyper-linear: params[b,o] = 2*(feat[b].lin_w[o] + lin_b[o]) --------------
__global__ void linear_kernel(const float* __restrict__ feat, const float* __restrict__ lw,
                              const float* __restrict__ lb, float* __restrict__ params) {
    int t = blockIdx.x * blockDim.x + threadIdx.x;
    if (t >= B_N * 1667) return;
    int o = t % 1667;
    int b = t / 1667;
    const float* f = feat + b * 32;
    const float* w = lw + (size_t)o * 32;
    float s = lb[o];
    #pragma unroll
    for (int k = 0; k < 32; ++k) s += f[k] * w[k];
    params[t] = 2.f * s;
}

// ============================================================================================
// WMMA f32 16x16x4 per-sample MLP.
// Weights are staged once per block into LDS in zero-padded, pre-transposed B layout:
//   wpad[k][n] (Kpad x NPAD), wpad = w[n*K+k] for k<K && n<O, else 0.
// Biases staged zero-padded. The WMMA inner loops are then fully straight-line:
// no guards, no EXEC manipulation, EXEC all-ones as the ISA requires.
// A frag: lane L -> row L&15, K = ks*4 + (L>=16?2:0) + {0,1}.
// B frag: lane L -> col L&15, same K mapping.  C/D: lane L, vgpr v -> row v+(L>=16?8:0), col L&15.
// ============================================================================================

template<int K, int O, int NPAD>
__device__ __forceinline__ void stage_w(const float* __restrict__ gp, float* __restrict__ dst,
                                        int lane) {
    constexpr int KPAD = (K + 3) & ~3;
    for (int idx = lane; idx < KPAD * NPAD; idx += 32) {
        int k = idx / NPAD, n = idx % NPAD;
        dst[idx] = (k < K && n < O) ? gp[n * K + k] : 0.f;
    }
}

template<int O, int NPAD>
__device__ __forceinline__ void stage_b(const float* __restrict__ gp, float* __restrict__ dst,
                                        int lane) {
    if (lane < NPAD) dst[lane] = (lane < O) ? gp[lane] : 0.f;
}

// One padded layer: act[16][32] -> act[16][32], relu. wpad is [Kpad][32], bpad is [32].
template<int K>
__device__ __forceinline__ void wmma_layer(const float* __restrict__ actIn,
                                           float* __restrict__ actOut,
                                           const float* __restrict__ wpad,
                                           const float* __restrict__ bpad, int lane) {
    constexpr int K4 = (K + 3) / 4;
    const int half = lane >> 4;
    const int nl = lane & 15;
    v2f a[K4];
    #pragma unroll
    for (int ks = 0; ks < K4; ++ks) {
        int k0 = ks * 4 + half * 2;
        a[ks][0] = actIn[nl * 32 + k0];
        a[ks][1] = actIn[nl * 32 + k0 + 1];
    }
    #pragma unroll
    for (int nt = 0; nt < 2; ++nt) {
        int n = nt * 16 + nl;
        float bv = bpad[n];
        v8f c;
        #pragma unroll
        for (int v = 0; v < 8; ++v) c[v] = bv;
        #pragma unroll
        for (int ks = 0; ks < K4; ++ks) {
            int k0 = ks * 4 + half * 2;
            v2f bf;
            bf[0] = wpad[(k0)     * 32 + n];
            bf[1] = wpad[(k0 + 1) * 32 + n];
            c = __builtin_amdgcn_wmma_f32_16x16x4_f32(false, a[ks], false, bf,
                                                      (short)0, c, false, false);
        }
        #pragma unroll
        for (int v = 0; v < 8; ++v) {
            int m = v + half * 8;
            actOut[m * 32 + n] = fmaxf(c[v], 0.f);
        }
    }
}

// Final layer: K=8, O=3 (one 16-wide N tile, NPAD=16), sigmoid, write [B,H,W,3].
__device__ __forceinline__ void wmma_final(const float* __restrict__ actIn,
                                           const float* __restrict__ wpad,
                                           const float* __restrict__ bpad,
                                           float* __restrict__ out, int b, int m0, int lane) {
    const int half = lane >> 4;
    const int nl = lane & 15;
    v2f a[2];
    #pragma unroll
    for (int ks = 0; ks < 2; ++ks) {
        int k0 = ks * 4 + half * 2;
        a[ks][0] = actIn[nl * 32 + k0];
        a[ks][1] = actIn[nl * 32 + k0 + 1];
    }
    const int n = nl;
    float bv = bpad[n];
    v8f c;
    #pragma unroll
    for (int v = 0; v < 8; ++v) c[v] = bv;
    #pragma unroll
    for (int ks = 0; ks < 2; ++ks) {
        int k0 = ks * 4 + half * 2;
        v2f bf;
        bf[0] = wpad[(k0)     * 16 + n];
        bf[1] = wpad[(k0 + 1) * 16 + n];
        c = __builtin_amdgcn_wmma_f32_16x16x4_f32(false, a[ks], false, bf,
                                                  (short)0, c, false, false);
    }
    if (n < 3) {
        #pragma unroll
        for (int v = 0; v < 8; ++v) {
            int m = v + half * 8;
            int point = m0 + m;
            out[((size_t)b * 1024 + point) * 3 + n] = 1.f / (1.f + expf(-c[v]));
        }
    }
}

// LDS layout for padded weights/biases (floats):
//  W0 [4 x32] @0      W1 [8 x32] @128    W2 [32x32] @384
//  W3 [32x32] @1408   W4 [8 x16] @2432
//  b0 @2560  b1 @2592  b2 @2624  b3 @2656  b4 @2688 (16)   total 2704
#define W0_OFF 0
#define W1_OFF 128
#define W2_OFF 384
#define W3_OFF 1408
#define W4_OFF 2432
#define B0_OFF 2560
#define B1_OFF 2592
#define B2_OFF 2624
#define B3_OFF 2656
#define B4_OFF 2688
#define SWB_SZ 2704

// One wave (32 threads) per (sample, 16-point tile). 6400 blocks total.
__global__ __launch_bounds__(32) void mlp_kernel(const float* __restrict__ x,
                                                 const float* __restrict__ params,
                                                 float* __restrict__ out) {
    __shared__ float swb[SWB_SZ];
    __shared__ float actA[16 * 32];
    __shared__ float actB[16 * 32];
    const int lane = threadIdx.x;
    const int blk = blockIdx.x;
    const int b = blk >> 6;
    const int m0 = (blk & 63) * 16;

    // stage zero-padded, transposed weights + padded biases
    // param offsets: w0@0 b0@24 w1@32 b1@288 w2@320 b2@1344 w3@1376 b3@1632 w4@1640 b4@1664
    const float* pp = params + (size_t)b * 1667;
    stage_w<3, 8, 32>(pp + 0,    swb + W0_OFF, lane);
    stage_w<8, 32, 32>(pp + 32,  swb + W1_OFF, lane);
    stage_w<32, 32, 32>(pp + 320, swb + W2_OFF, lane);
    stage_w<32, 8, 32>(pp + 1376, swb + W3_OFF, lane);
    stage_w<8, 3, 16>(pp + 1640, swb + W4_OFF, lane);
    stage_b<8, 32>(pp + 24,   swb + B0_OFF, lane);
    stage_b<32, 32>(pp + 288, swb + B1_OFF, lane);
    stage_b<32, 32>(pp + 1344, swb + B2_OFF, lane);
    stage_b<8, 32>(pp + 1632, swb + B3_OFF, lane);
    stage_b<3, 16>(pp + 1664, swb + B4_OFF, lane);

    // z0[m] = [ i/32, j/32, x[b,0,i,j] ], zero-padded to 32 cols
    for (int idx = lane; idx < 512; idx += 32) {
        int m = idx >> 5, k = idx & 31;
        int point = m0 + m;
        float v = 0.f;
        if (k == 0)      v = (float)(point >> 5) * (1.f / 32.f);
        else if (k == 1) v = (float)(point & 31) * (1.f / 32.f);
        else if (k == 2) v = x[(size_t)b * 1024 + point];
        actA[idx] = v;
    }
    __syncthreads();

    wmma_layer<3 >(actA, actB, swb + W0_OFF, swb + B0_OFF, lane); __syncthreads();
    wmma_layer<8 >(actB, actA, swb + W1_OFF, swb + B1_OFF, lane); __syncthreads();
    wmma_layer<32>(actA, actB, swb + W2_OFF, swb + B2_OFF, lane); __syncthreads();
    wmma_layer<32>(actB, actA, swb + W3_OFF, swb + B3_OFF, lane); __syncthreads();
    wmma_final(actA, swb + W4_OFF, swb + B4_OFF, out, b, m0, lane);
}

extern "C" void kernel_launch(void* const* d_in, const int* in_sizes, int n_in,
                              void* d_out, int out_size, void* d_ws, size_t ws_size,
                              hipStream_t stream) {
    const float* x   = (const float*)d_in[0];
    const float* c1w = (const float*)d_in[1];
    const float* c1b = (const float*)d_in[2];
    const float* g1  = (const float*)d_in[3];
    const float* be1 = (const float*)d_in[4];
    const float* c2w = (const float*)d_in[5];
    const float* c2b = (const float*)d_in[6];
    const float* g2  = (const float*)d_in[7];
    const float* be2 = (const float*)d_in[8];
    const float* c3w = (const float*)d_in[9];
    const float* c3b = (const float*)d_in[10];
    const float* g3  = (const float*)d_in[11];
    const float* be3 = (const float*)d_in[12];
    const float* c4w = (const float*)d_in[13];
    const float* c4b = (const float*)d_in[14];
    const float* lw  = (const float*)d_in[15];
    const float* lb  = (const float*)d_in[16];

    float* ws  = (float*)d_ws;
    float* c1o  = ws;               // [B,6,32,32]  614400
    float* h1   = c1o + 614400;     // [B,6,16,16]  153600
    float* c2o  = h1  + 153600;     // [B,20,16,16] 512000
    float* h2   = c2o + 512000;     // [B,20,8,8]   128000
    float* c3o  = h2  + 128000;     // [B,4,8,8]    25600
    float* h3   = c3o + 25600;      // [B,4,4,4]    6400
    float* feat = h3  + 6400;       // [B,32]       3200
    float* par  = feat + 3200;      // [B,1667]     166700
    float* st1  = par + 166700;     // 12
    float* st2  = st1 + 12;         // 40
    float* st3  = st2 + 40;         // 8
    float* out  = (float*)d_out;    // [B,32,32,3]  307200

    convk<1, 6, 32><<<(B_N * 6 * 1024 + 255) / 256, 256, 0, stream>>>(x, c1w, c1b, c1o);
    stats_kernel<6, 32><<<6, 256, 0, stream>>>(c1o, st1);
    bnpool_kernel<6, 32><<<(B_N * 6 * 256 + 255) / 256, 256, 0, stream>>>(c1o, st1, g1, be1, h1);

    convk<6, 20, 16><<<(B_N * 20 * 256 + 255) / 256, 256, 0, stream>>>(h1, c2w, c2b, c2o);
    stats_kernel<20, 16><<<20, 256, 0, stream>>>(c2o, st2);
    bnpool_kernel<20, 16><<<(B_N * 20 * 64 + 255) / 256, 256, 0, stream>>>(c2o, st2, g2, be2, h2);

    convk<20, 4, 8><<<(B_N * 4 * 64 + 255) / 256, 256, 0, stream>>>(h2, c3w, c3b, c3o);
    stats_kernel<4, 8><<<4, 256, 0, stream>>>(c3o, st3);
    bnpool_kernel<4, 8><<<(B_N * 4 * 16 + 255) / 256, 256, 0, stream>>>(c3o, st3, g3, be3, h3);

    conv4_kernel<<<(B_N * 32 + 255) / 256, 256, 0, stream>>>(h3, c4w, c4b, feat);
    linear_kernel<<<(B_N * 1667 + 255) / 256, 256, 0, stream>>>(feat, lw, lb, par);

    mlp_kernel<<<B_N * 64, 32, 0, stream>>>(x, par, out);
}